// GraphAdjNetwork_20126216749872
// MI455X (gfx1250) — compile-verified
//
#include <hip/hip_runtime.h>

typedef float v2f __attribute__((ext_vector_type(2)));
typedef float v8f __attribute__((ext_vector_type(8)));

#define NN 768
#define CC 256

// Kernel 1: s = x @ w via V_WMMA_F32_16X16X4_F32 (full fp32 path).
// One wave (32 threads) per 16-row tile: 64 accumulating WMMAs over K=256.
// B has w[k..k+3] replicated into all 16 columns, so D[i][j] == s_{r0+i} for all j.
__global__ __launch_bounds__(32) void s_wmma_kernel(const float* __restrict__ x,
                                                    const float* __restrict__ w,
                                                    float* __restrict__ s) {
    const int lane  = threadIdx.x;        // 0..31, wave32
    const int r0    = blockIdx.x * 16;    // row-tile base
    const int m     = lane & 15;          // A: lane = M row;  B: lane = N col
    const int khalf = (lane >> 4) * 2;    // lanes 0-15 -> K{0,1}, lanes 16-31 -> K{2,3}

    const float* xrow = x + (r0 + m) * CC + khalf;
    const float* wp   = w + khalf;

    v8f c = {0.f, 0.f, 0.f, 0.f, 0.f, 0.f, 0.f, 0.f};
    for (int k = 0; k < CC; k += 4) {
        v2f a, b;
        a.x = xrow[k];      // A VGPR0: K=0 (lanes 0-15) / K=2 (lanes 16-31)
        a.y = xrow[k + 1];  // A VGPR1: K=1 / K=3
        b.x = wp[k];        // B VGPR0: same K striping, identical across N lanes
        b.y = wp[k + 1];
        c = __builtin_amdgcn_wmma_f32_16x16x4_f32(
                /*neg_a=*/false, a, /*neg_b=*/false, b,
                /*c_mod=*/(short)0, c, /*reuse_a=*/false, /*reuse_b=*/false);
    }

    // D layout: lanes 0-15 VGPR v = D[v][lane]; lanes 16-31 VGPR v = D[v+8][lane-16].
    // Column 0 (lane 0) and column 0 of the upper half (lane 16) carry all 16 rows.
    if (m == 0) {
        const int base = r0 + (lane >> 4) * 8;
        #pragma unroll
        for (int v = 0; v < 8; ++v) s[base + v] = c[v];
    }
}

// Kernel 2: streaming output writer. out[i][j] = relu((s_i - s_j)*scale + c0)/N.
// One float4 per thread: coalesced b128 load of s[j..j+3], coalesced b128 store.
__global__ __launch_bounds__(256) void adj_kernel(const float* __restrict__ s,
                                                  const float* __restrict__ gamma,
                                                  const float* __restrict__ beta,
                                                  const float* __restrict__ rmean,
                                                  const float* __restrict__ rvar,
                                                  float* __restrict__ out) {
    const int idx = blockIdx.x * 256 + threadIdx.x;   // 0 .. 768*192-1
    const int i   = idx / 192;                         // row
    const int j4  = (idx - i * 192) * 4;               // column group

    const float scale = gamma[0] / sqrtf(rvar[0] + 1e-5f);
    const float c0    = beta[0] - rmean[0] * scale;
    const float invN  = 1.0f / (float)NN;

    const float  si = s[i];
    const float4 sj = *reinterpret_cast<const float4*>(s + j4);

    float4 r;
    r.x = fmaxf(fmaf(si - sj.x, scale, c0), 0.0f) * invN;
    r.y = fmaxf(fmaf(si - sj.y, scale, c0), 0.0f) * invN;
    r.z = fmaxf(fmaf(si - sj.z, scale, c0), 0.0f) * invN;
    r.w = fmaxf(fmaf(si - sj.w, scale, c0), 0.0f) * invN;

    *reinterpret_cast<float4*>(out + idx * 4) = r;    // == &out[i*768 + j4]
}

extern "C" void kernel_launch(void* const* d_in, const int* in_sizes, int n_in,
                              void* d_out, int out_size, void* d_ws, size_t ws_size,
                              hipStream_t stream) {
    const float* x     = (const float*)d_in[0];  // (1, 768, 256)
    const float* w     = (const float*)d_in[1];  // (256,)
    const float* gamma = (const float*)d_in[2];  // (1,)
    const float* beta  = (const float*)d_in[3];  // (1,)
    const float* rmean = (const float*)d_in[4];  // (1,)
    const float* rvar  = (const float*)d_in[5];  // (1,)
    float*       out   = (float*)d_out;          // (768, 768)
    float*       s     = (float*)d_ws;           // 768 floats of scratch

    s_wmma_kernel<<<NN / 16, 32, 0, stream>>>(x, w, s);
    adj_kernel<<<(NN * (NN / 4)) / 256, 256, 0, stream>>>(s, gamma, beta, rmean, rvar, out);
}